// GRUNet_24386824306758
// MI455X (gfx1250) — compile-verified
//
#include <hip/hip_runtime.h>
#include <hip/hip_bf16.h>
#include <stdint.h>
#include <math.h>

// Problem constants (from reference setup_inputs)
#define S_LEN 8192
#define HID   1024
#define G3    3072   // 3*HID
#define OUTD  512
#define NBLK  32     // persistent scan blocks
#define AST   20     // LDS row stride (floats): 16 data + 4 pad -> conflict-free,
                     // keeps every 4-float k-quad 16B aligned for async B128 copies

typedef float v2f __attribute__((ext_vector_type(2)));
typedef float v8f __attribute__((ext_vector_type(8)));
typedef int   v4i __attribute__((ext_vector_type(4)));
typedef int   v2i __attribute__((ext_vector_type(2)));

// ---------------------------------------------------------------------------
// CDNA5 async global->LDS copies (ASYNCcnt-tracked, bypass VGPRs).
// Guarded so the build falls back to sync loads if the toolchain lacks them.
// Param types per hipcc diagnostic: vector-of-int pointers, global src (AS1),
// LDS dst (AS3).
// ---------------------------------------------------------------------------
#if __has_builtin(__builtin_amdgcn_global_load_async_to_lds_b128) && \
    __has_builtin(__builtin_amdgcn_global_load_async_to_lds_b64)  && \
    __has_builtin(__builtin_amdgcn_s_wait_asynccnt)
#define USE_ASYNC 1
#else
#define USE_ASYNC 0
#endif

#if USE_ASYNC
typedef __attribute__((address_space(1))) v4i gv4i;
typedef __attribute__((address_space(3))) v4i lv4i;
typedef __attribute__((address_space(1))) v2i gv2i;
typedef __attribute__((address_space(3))) v2i lv2i;

__device__ __forceinline__ void async_ld_b128(const float* g, float* l) {
  __builtin_amdgcn_global_load_async_to_lds_b128(
      (gv4i*)(uintptr_t)g, (lv4i*)(uint32_t)(uintptr_t)l, 0, 0);
}
__device__ __forceinline__ void async_ld_b64(const float* g, float* l) {
  __builtin_amdgcn_global_load_async_to_lds_b64(
      (gv2i*)(uintptr_t)g, (lv2i*)(uint32_t)(uintptr_t)l, 0, 0);
}
#endif

// ---------------------------------------------------------------------------
// GEMM with bias: C[M x N] = A[M x K] @ W[N x K]^T + bias[N]
// 256 threads = 8 waves; block tile 32(M) x 128(N); K chunks of 16.
// Each wave computes a 16x32 C tile (2 accumulators sharing the A fragment)
// with V_WMMA_F32_16X16X4_F32. LDS is double-buffered; chunk i+1 streams in
// via GLOBAL_LOAD_ASYNC_TO_LDS while chunk i's WMMAs execute.
// Requires: M % 32 == 0, N % 128 == 0, K % 16 == 0.
// ---------------------------------------------------------------------------
__global__ __launch_bounds__(256)
void wmma_gemm_bias(const float* __restrict__ A, const float* __restrict__ W,
                    const float* __restrict__ bias, float* __restrict__ C,
                    int K, int N) {
  __shared__ float As[2][32 * AST];    // [m][k]
  __shared__ float Bs[2][128 * AST];   // [n][k]

  const int tid   = threadIdx.x;
  const int lane  = tid & 31;
  const int wid   = tid >> 5;     // wave 0..7
  const int idx   = lane & 15;
  const int grp   = lane >> 4;    // half-wave group
  const int wm    = wid & 1;      // 2 M sub-tiles
  const int wn    = wid >> 1;     // 4 N sub-tiles of 32 cols
  const int tileM = blockIdx.x * 32;
  const int tileN = blockIdx.y * 128;

  // staging assignment: 512 B-quads (128 rows x 4 quads) + 256 A-pairs
  const int brow0 = tid >> 2,         bkq0 = (tid & 3) * 4;
  const int brow1 = (tid + 256) >> 2, bkq1 = (tid & 3) * 4;
  const int am    = tid >> 3,         ako  = (tid & 7) * 2;

  v8f acc0 = {0.f,0.f,0.f,0.f,0.f,0.f,0.f,0.f};
  v8f acc1 = {0.f,0.f,0.f,0.f,0.f,0.f,0.f,0.f};

  auto stage = [&](int kc, int b) {
#if USE_ASYNC
    async_ld_b128(W + (size_t)(tileN + brow0) * K + kc + bkq0, &Bs[b][brow0 * AST + bkq0]);
    async_ld_b128(W + (size_t)(tileN + brow1) * K + kc + bkq1, &Bs[b][brow1 * AST + bkq1]);
    async_ld_b64 (A + (size_t)(tileM + am)    * K + kc + ako,  &As[b][am    * AST + ako]);
#else
    *(float4*)&Bs[b][brow0 * AST + bkq0] =
        *(const float4*)(W + (size_t)(tileN + brow0) * K + kc + bkq0);
    *(float4*)&Bs[b][brow1 * AST + bkq1] =
        *(const float4*)(W + (size_t)(tileN + brow1) * K + kc + bkq1);
    *(float2*)&As[b][am * AST + ako] =
        *(const float2*)(A + (size_t)(tileM + am) * K + kc + ako);
#endif
  };

  const int nch = K >> 4;
  stage(0, 0);

  for (int i = 0; i < nch; ++i) {
    const int  cur  = i & 1;
    const bool more = (i + 1 < nch);
    if (more) stage((i + 1) << 4, cur ^ 1);     // prefetch next chunk (3 asyncs)
#if USE_ASYNC
    // async copies complete in order: leaving <=3 in flight means the current
    // chunk's 3 copies have landed in LDS.
    if (more) __builtin_amdgcn_s_wait_asynccnt(3);
    else      __builtin_amdgcn_s_wait_asynccnt(0);
#endif
    __syncthreads();

    // Fragment layouts per ISA 7.12.2 (32-bit A/B):
    //   A: lanes 0-15 hold K=k0,k0+1 (v0,v1); lanes 16-31 hold K=k0+2,k0+3
    const float* Arow  = &As[cur][(wm * 16 + idx) * AST];
    const float* Brow0 = &Bs[cur][(wn * 32 + idx) * AST];
    const float* Brow1 = Brow0 + 16 * AST;
    #pragma unroll
    for (int k0 = 0; k0 < 16; k0 += 4) {
      const int ka = k0 + 2 * grp;
      v2f a  = { Arow[ka],  Arow[ka + 1] };
      v2f b0 = { Brow0[ka], Brow0[ka + 1] };
      v2f b1 = { Brow1[ka], Brow1[ka + 1] };
      acc0 = __builtin_amdgcn_wmma_f32_16x16x4_f32(
                 false, a, false, b0, (short)0, acc0, false, false);
      acc1 = __builtin_amdgcn_wmma_f32_16x16x4_f32(
                 false, a, false, b1, (short)0, acc1, false, false);
    }
    __syncthreads();   // all waves done reading buf[cur] before it is refilled
  }

  // Writeout: C layout per ISA — VGPR r: lanes 0-15 M=r, lanes 16-31 M=r+8
  const int col0 = tileN + wn * 32 + idx;
  const int col1 = col0 + 16;
  const float bv0 = bias[col0];
  const float bv1 = bias[col1];
  #pragma unroll
  for (int r = 0; r < 8; ++r) {
    const int row = tileM + wm * 16 + r + 8 * grp;
    C[(size_t)row * N + col0] = acc0[r] + bv0;
    C[(size_t)row * N + col1] = acc1[r] + bv1;
  }
}

// ---------------------------------------------------------------------------
// Grid-wide sense-reversal barrier (state in workspace, re-init each launch)
// ---------------------------------------------------------------------------
__device__ __forceinline__ void grid_barrier(unsigned* counter,
                                             volatile unsigned* gen,
                                             unsigned nblk) {
  __syncthreads();
  if (threadIdx.x == 0) {
    __threadfence();
    const unsigned g = *gen;
    if (atomicAdd(counter, 1u) == nblk - 1u) {
      *(volatile unsigned*)counter = 0u;   // reset for next phase
      __threadfence();
      atomicAdd((unsigned*)gen, 1u);       // release everyone
    } else {
      while (*gen == g) __builtin_amdgcn_s_sleep(2);
    }
    __threadfence();
  }
  __syncthreads();
}

__global__ void init_bar(unsigned* bar) {
  if (threadIdx.x < 2) bar[threadIdx.x] = 0u;
}

// ---------------------------------------------------------------------------
// Persistent GRU scan: one grid barrier per timestep.
// 32 blocks x 256 threads = 256 waves; each wave owns 4 hidden units and
// computes all three gate dot products (rows j, j+H, j+2H) for its units.
// W_hh (12 MB) is served from L2 after the first step.
// ---------------------------------------------------------------------------
__global__ __launch_bounds__(256)
void gru_scan(const float* __restrict__ Whh, const float* __restrict__ bhh,
              const float* __restrict__ gi, const float* __restrict__ h0,
              float* __restrict__ ys, float* __restrict__ hT,
              float* hcur, unsigned* bar) {
  __shared__ float hl[HID];
  const int tid   = threadIdx.x;
  const int lane  = tid & 31;
  const int wid   = tid >> 5;
  const int waveG = blockIdx.x * 8 + wid;    // 0..255
  const int gtid  = blockIdx.x * 256 + tid;  // 0..8191

  if (gtid < HID) hcur[gtid] = h0[gtid];
  grid_barrier(bar, bar + 1, NBLK);

  for (int t = 0; t < S_LEN; ++t) {
    // broadcast h into LDS (4 KB, float4 coalesced)
    ((float4*)hl)[tid] = ((const float4*)hcur)[tid];
    __syncthreads();

    const float* git = gi + (size_t)t * G3;
    #pragma unroll
    for (int u = 0; u < 4; ++u) {
      const int j = waveG * 4 + u;
      const float* wr = Whh + (size_t)j * HID;
      const float* wz = Whh + (size_t)(j + HID) * HID;
      const float* wn = Whh + (size_t)(j + 2 * HID) * HID;
      float sR = 0.f, sZ = 0.f, sN = 0.f;
      for (int k = lane; k < HID; k += 32) {   // lane-coalesced weight reads
        const float hv = hl[k];
        sR = fmaf(wr[k], hv, sR);
        sZ = fmaf(wz[k], hv, sZ);
        sN = fmaf(wn[k], hv, sN);
      }
      #pragma unroll
      for (int m = 16; m >= 1; m >>= 1) {      // wave32 reduction
        sR += __shfl_xor(sR, m, 32);
        sZ += __shfl_xor(sZ, m, 32);
        sN += __shfl_xor(sN, m, 32);
      }
      if (lane == 0) {
        const float gr = sR + bhh[j];
        const float gz = sZ + bhh[j + HID];
        const float gn = sN + bhh[j + 2 * HID];
        const float r  = 1.f / (1.f + __expf(-(git[j] + gr)));
        const float z  = 1.f / (1.f + __expf(-(git[j + HID] + gz)));
        const float n  = tanhf(git[j + 2 * HID] + r * gn);
        const float hnew = (1.f - z) * n + z * hl[j];
        hcur[j] = hnew;
        ys[(size_t)t * HID + j] = hnew;
      }
    }
    grid_barrier(bar, bar + 1, NBLK);   // make hcur globally visible
  }

  if (gtid < HID) hT[gtid] = hcur[gtid];
}

// ---------------------------------------------------------------------------
// Launch: init -> 3x (gi GEMM -> scan) -> FC GEMM
// ---------------------------------------------------------------------------
extern "C" void kernel_launch(void* const* d_in, const int* in_sizes, int n_in,
                              void* d_out, int out_size, void* d_ws, size_t ws_size,
                              hipStream_t stream) {
  const float* x   = (const float*)d_in[0];
  const float* h0  = (const float*)d_in[1];
  const float* Wih = (const float*)d_in[2];
  const float* Whh = (const float*)d_in[3];
  const float* bih = (const float*)d_in[4];
  const float* bhh = (const float*)d_in[5];
  const float* fcw = (const float*)d_in[6];
  const float* fcb = (const float*)d_in[7];
  float* out = (float*)d_out;

  // workspace layout (floats)
  float* ws   = (float*)d_ws;
  float* gi   = ws;                                   // S*3H  (96 MB)
  float* buf0 = gi   + (size_t)S_LEN * G3;            // S*H   (32 MB)
  float* buf1 = buf0 + (size_t)S_LEN * HID;           // S*H   (32 MB)
  float* hcur = buf1 + (size_t)S_LEN * HID;           // H
  unsigned* bar = (unsigned*)(hcur + HID);            // 2 words

  init_bar<<<1, 32, 0, stream>>>(bar);

  const float* lin = x;
  float* lout = buf0;
  for (int l = 0; l < 3; ++l) {
    dim3 gg(S_LEN / 32, G3 / 128);
    wmma_gemm_bias<<<gg, 256, 0, stream>>>(
        lin, Wih + (size_t)l * G3 * HID, bih + (size_t)l * G3, gi, HID, G3);
    gru_scan<<<NBLK, 256, 0, stream>>>(
        Whh + (size_t)l * G3 * HID, bhh + (size_t)l * G3, gi,
        h0 + (size_t)l * HID, lout,
        out + (size_t)S_LEN * OUTD + (size_t)l * HID,  // h_out tail of d_out
        hcur, bar);
    lin  = lout;
    lout = (l == 0) ? buf1 : buf0;
  }

  // final FC: y = lin @ fc_w^T + fc_b   (lin == buf0 after 3 layers)
  dim3 gf(S_LEN / 32, OUTD / 128);
  wmma_gemm_bias<<<gf, 256, 0, stream>>>(lin, fcw, fcb, out, HID, OUTD);
}